// SpikeLN_35330400976906
// MI455X (gfx1250) — compile-verified
//
#include <hip/hip_runtime.h>
#include <cstdint>

// SpikeLN fused kernel for MI455X (gfx1250).
// out = (x - mean_row(x)) * mlp(var_row(x)) * gamma + beta
// HBM-bound (402 MB @ 23.3 TB/s ~= 17us). One wave32 per 768-col row;
// async global->LDS double-buffered prefetch (ASYNCcnt path); wave-level
// sum / sum-of-squares reductions done on the matrix unit via
// V_WMMA_F32_16X16X4_F32 with a ones B-operand (result broadcast to all lanes).

#define N_COLS    768
#define F4_ROW    (N_COLS / 4)     // 192 float4 per row
#define F4_LANE   (F4_ROW / 32)    // 6 float4 per lane
#define WPB       8                // waves per block (256 threads)
#define ROW_BYTES (N_COLS * 4)     // 3072

typedef __attribute__((ext_vector_type(2))) float v2f;
typedef __attribute__((ext_vector_type(8))) float v8f;

__device__ __forceinline__ uint32_t lds_byte_off(const void* p) {
  // generic -> AS(3) cast yields the workgroup-relative LDS byte offset,
  // which is what the async-load VDST operand expects (HW adds LDS_BASE).
  return (uint32_t)(uintptr_t)(const __attribute__((address_space(3))) void*)p;
}

__device__ __forceinline__ void async_row_prefetch(const float* gRow,
                                                   uint32_t ldsOff, int lane) {
  // Each lane copies 6 x 16B chunks, 512B apart (coalesced across the wave).
  // The instruction IOFFSET is added to BOTH the global and LDS addresses
  // (ISA 15.18.3 async pseudocode): one address pair + literal offsets
  // covers the whole row. 6 instructions, ASYNCcnt += 6.
  uint64_t ga = (uint64_t)(uintptr_t)gRow + (uint64_t)((uint32_t)lane * 16u);
  uint32_t la = ldsOff + (uint32_t)lane * 16u;
#define ASYNC_LD16(OFF)                                                  \
  asm volatile("global_load_async_to_lds_b128 %0, %1, off offset:" #OFF  \
               :: "v"(la), "v"(ga) : "memory")
  ASYNC_LD16(0);
  ASYNC_LD16(512);
  ASYNC_LD16(1024);
  ASYNC_LD16(1536);
  ASYNC_LD16(2048);
  ASYNC_LD16(2560);
#undef ASYNC_LD16
}

// Full 32-lane all-reduce (sum) on the matrix unit.
// A-matrix f32 16x4 layout: lanes 0-15 hold M=lane {v0:K0, v1:K1},
// lanes 16-31 hold M=lane-16 {v0:K2, v1:K3}.  C/D: d[i] = row M=i
// (lanes 0-15) / row M=i+8 (lanes 16-31), identical across columns.
// Step 1: A = (s,s), B = ones  =>  row m of D = t_m = 2*(s_m + s_{m+16}).
// Step 2: 4 accumulating WMMAs, A = (d[2j], d[2j+1]) => each row-sum adds
// t_{2j}+t_{2j+1}+t_{2j+8}+t_{2j+9}; over j=0..3 covers all 16 t exactly
// once => acc = 2 * sum_{l=0..31} s_l, broadcast to every lane/element.
__device__ __forceinline__ float wmma_allreduce32(float s) {
  v2f ones; ones[0] = 1.0f; ones[1] = 1.0f;
  v2f a;    a[0]    = s;    a[1]    = s;
  v8f zc = {};
  v8f t = __builtin_amdgcn_wmma_f32_16x16x4_f32(
      false, a, false, ones, (short)0, zc, false, false);
  v8f acc = {};
#pragma unroll
  for (int j = 0; j < 4; ++j) {
    v2f aj; aj[0] = t[2 * j]; aj[1] = t[2 * j + 1];
    acc = __builtin_amdgcn_wmma_f32_16x16x4_f32(
        false, aj, false, ones, (short)0, acc, false, false);
  }
  return acc[0] * 0.5f;
}

__global__ __launch_bounds__(WPB * 32) void spikeln_fused_kernel(
    const float* __restrict__ x,
    const float* __restrict__ w1, const float* __restrict__ b1,
    const float* __restrict__ w2, const float* __restrict__ b2,
    const float* __restrict__ gamma, const float* __restrict__ beta,
    float* __restrict__ out, int nRows)
{
  __shared__ float4 lds4[WPB * 2 * F4_ROW];   // 8 waves * 2 bufs * 3KB = 48KB

  const int lane = threadIdx.x & 31;
  const int wave = (int)threadIdx.x >> 5;
  const int gridWaves = (int)gridDim.x * WPB;

  // Tiny distilled-rsqrt MLP params: wave-uniform -> scalar loads, registers.
  float W1[8], B1v[8], W2[8];
#pragma unroll
  for (int i = 0; i < 8; ++i) { W1[i] = w1[i]; B1v[i] = b1[i]; W2[i] = w2[i]; }
  const float B2 = b2[0];

  // gamma/beta for this lane's fixed column set (same every row): cache once.
  float4 G[F4_LANE], Bt[F4_LANE];
#pragma unroll
  for (int j = 0; j < F4_LANE; ++j) {
    G[j]  = ((const float4*)gamma)[j * 32 + lane];
    Bt[j] = ((const float4*)beta )[j * 32 + lane];
  }

  const uint32_t ldsOff0 = lds_byte_off(&lds4[wave * 2 * F4_ROW]);

  int row = (int)blockIdx.x * WPB + wave;   // wave-uniform => EXEC all-ones
  int cb = 0;
  if (row < nRows)
    async_row_prefetch(x + (size_t)row * N_COLS, ldsOff0, lane);

  while (row < nRows) {
    const int nxt = row + gridWaves;
    if (nxt < nRows) {
      // Issue next row's prefetch before waiting on the current one:
      // 12 in flight -> wait ASYNCcnt<=6 (async loads complete in order).
      async_row_prefetch(x + (size_t)nxt * N_COLS,
                         ldsOff0 + (uint32_t)((cb ^ 1) * ROW_BYTES), lane);
      asm volatile("s_wait_asynccnt 0x6" ::: "memory");
    } else {
      asm volatile("s_wait_asynccnt 0x0" ::: "memory");
    }

    const float4* buf = &lds4[(wave * 2 + cb) * F4_ROW];
    float4 v[F4_LANE];
#pragma unroll
    for (int j = 0; j < F4_LANE; ++j) v[j] = buf[j * 32 + lane];

    // mean over row (WMMA all-reduce of per-lane partial sums)
    float s = 0.f;
#pragma unroll
    for (int j = 0; j < F4_LANE; ++j)
      s += (v[j].x + v[j].y) + (v[j].z + v[j].w);
    const float mean = wmma_allreduce32(s) * (1.0f / (float)N_COLS);

    // centered values + variance (matches reference: var of centered row)
    float q = 0.f;
#pragma unroll
    for (int j = 0; j < F4_LANE; ++j) {
      v[j].x -= mean; v[j].y -= mean; v[j].z -= mean; v[j].w -= mean;
      q += v[j].x * v[j].x + v[j].y * v[j].y;
      q += v[j].z * v[j].z + v[j].w * v[j].w;
    }
    const float var = wmma_allreduce32(q) * (1.0f / (float)N_COLS);

    // distilled sqrt-inverse MLP: h = relu(var*w1 + b1); rec = h.w2 + b2
    float rec = B2;
#pragma unroll
    for (int i = 0; i < 8; ++i)
      rec += fmaxf(fmaf(var, W1[i], B1v[i]), 0.f) * W2[i];

    // X2X encoder relu(p)-relu(-p) == p (exact identity); apply affine + store
    float4* orow = (float4*)out + (size_t)row * F4_ROW;
#pragma unroll
    for (int j = 0; j < F4_LANE; ++j) {
      float4 r;
      r.x = fmaf(v[j].x * rec, G[j].x, Bt[j].x);
      r.y = fmaf(v[j].y * rec, G[j].y, Bt[j].y);
      r.z = fmaf(v[j].z * rec, G[j].z, Bt[j].z);
      r.w = fmaf(v[j].w * rec, G[j].w, Bt[j].w);
      orow[j * 32 + lane] = r;
    }

    row = nxt;
    cb ^= 1;
  }
}

extern "C" void kernel_launch(void* const* d_in, const int* in_sizes, int n_in,
                              void* d_out, int out_size, void* d_ws, size_t ws_size,
                              hipStream_t stream) {
  (void)n_in; (void)out_size; (void)d_ws; (void)ws_size;
  const float* x     = (const float*)d_in[0];
  const float* w1    = (const float*)d_in[1];
  const float* b1    = (const float*)d_in[2];
  const float* w2    = (const float*)d_in[3];
  const float* b2    = (const float*)d_in[4];
  const float* gamma = (const float*)d_in[5];
  const float* beta  = (const float*)d_in[6];
  float* out = (float*)d_out;

  const int nRows = in_sizes[0] / N_COLS;         // 65536 for [8,8192,768]
  const int rowsPerWave = 4;                      // grid-stride iterations
  int blocks = (nRows + WPB * rowsPerWave - 1) / (WPB * rowsPerWave);
  if (blocks < 1) blocks = 1;

  spikeln_fused_kernel<<<blocks, WPB * 32, 0, stream>>>(
      x, w1, b1, w2, b2, gamma, beta, out, nRows);
}